// Propagation_1228360646954
// MI455X (gfx1250) — compile-verified
//
#include <hip/hip_runtime.h>
#include <hip/hip_bf16.h>

// Propagation: out = (1-ALPHA) * segment_sum(vals * x[cols], rows) + ALPHA * h
// N_NODES=100000, N_EDGES=1600000, D_FEAT=50, ALPHA=0.1
//
// MI455X reasoning: working set ~66 MB < 192 MB L2 => everything L2-resident.
// ~160 MFLOP vs ~70 MB of traffic => pure data-movement/atomic problem; WMMA
// has no dense tile structure to exploit (random rows), so the gfx1250 paths
// that pay are: async-to-LDS edge staging (ASYNCcnt pipeline), global
// prefetch, and non-returning global_atomic_add_f32 resolved in L2.

#define D_FEAT   50
#define SCALE    0.9f   // (1 - ALPHA)
#define ALPHA_F  0.1f
#define TILE_E   512    // edges per tile (2 per lane via b64 async loads)
#define BLOCK    256

// ---------------------------------------------------------------------------
// Kernel: scatter-accumulate  out[row*50+f] += 0.9 * val * x[col*50+f]
// Edge triples staged into LDS with async b64 loads (2 edges/lane/array),
// double-buffered so the next tile's loads are in flight during compute.
// ---------------------------------------------------------------------------
__global__ void __launch_bounds__(BLOCK)
spmm_scatter_kernel(const float* __restrict__ x,
                    const float* __restrict__ vals,
                    const int*   __restrict__ rows,
                    const int*   __restrict__ cols,
                    float*       __restrict__ out,
                    int n_edges, int n_tiles)
{
    __shared__ float s_vals[2][TILE_E];
    __shared__ int   s_rows[2][TILE_E];
    __shared__ int   s_cols[2][TILE_E];

    const int tid = threadIdx.x;

    // 32-bit LDS byte addresses of this thread's 8-byte staging slots
    // (low 32 bits of flat shared address == wave-relative LDS offset)
    unsigned lv0 = (unsigned)(unsigned long long)&s_vals[0][tid * 2];
    unsigned lr0 = (unsigned)(unsigned long long)&s_rows[0][tid * 2];
    unsigned lc0 = (unsigned)(unsigned long long)&s_cols[0][tid * 2];
    unsigned lv1 = (unsigned)(unsigned long long)&s_vals[1][tid * 2];
    unsigned lr1 = (unsigned)(unsigned long long)&s_rows[1][tid * 2];
    unsigned lc1 = (unsigned)(unsigned long long)&s_cols[1][tid * 2];

    auto issue_tile = [&](int tile, int buf) {
        // this lane stages edges [e2, e2+1] of the tile (8 bytes per array)
        int e2 = tile * TILE_E + tid * 2;
        if (e2 > n_edges - 2) e2 = n_edges - 2;   // clamp; dup loads harmless
        unsigned off = (unsigned)e2 * 4u;         // byte offset
        unsigned dv = buf ? lv1 : lv0;
        unsigned dr = buf ? lr1 : lr0;
        unsigned dc = buf ? lc1 : lc0;
        asm volatile("global_load_async_to_lds_b64 %0, %1, %2"
                     :: "v"(dv), "v"(off), "s"(vals) : "memory");
        asm volatile("global_load_async_to_lds_b64 %0, %1, %2"
                     :: "v"(dr), "v"(off), "s"(rows) : "memory");
        asm volatile("global_load_async_to_lds_b64 %0, %1, %2"
                     :: "v"(dc), "v"(off), "s"(cols) : "memory");
    };

    // 250 active lanes: 5 edge-groups x 50 features (wave32-friendly, 97.6%)
    const int g = tid / D_FEAT;                 // 0..5 (g==5 idle in compute)
    const int f = tid - g * D_FEAT;             // 0..49

    int t = blockIdx.x;
    if (t >= n_tiles) return;

    int b = 0;
    issue_tile(t, 0);

    for (; t < n_tiles; t += gridDim.x) {
        const int t_next = t + gridDim.x;
        const bool have_next = (t_next < n_tiles);

        if (have_next) {
            issue_tile(t_next, b ^ 1);
            // L2 warm-up two tiles ahead (speculative; dropped if OOB anyway)
            long pe = (long)(t + 2 * (long)gridDim.x) * TILE_E + tid;
            if (pe < n_edges) {
                __builtin_prefetch(vals + pe, 0, 3);
                __builtin_prefetch(rows + pe, 0, 3);
                __builtin_prefetch(cols + pe, 0, 3);
            }
            // keep the 3 just-issued loads outstanding; wait for tile t's 3
            asm volatile("s_wait_asynccnt 0x3" ::: "memory");
        } else {
            asm volatile("s_wait_asynccnt 0x0" ::: "memory");
        }
        __syncthreads();   // all waves' tile-t triples now resident in buf b

        if (g < 5) {
            const int e_base = t * TILE_E;
            #pragma unroll 4
            for (int p = 0; p < TILE_E; p += 5) {
                const int e_local = p + g;
                if (e_local >= TILE_E) break;
                const int e = e_base + e_local;
                if (e < n_edges) {
                    const float v = s_vals[b][e_local];
                    const int   r = s_rows[b][e_local];
                    const int   c = s_cols[b][e_local];
                    const float xv = x[c * D_FEAT + f];  // coalesced gather (L2)
                    atomicAdd(&out[r * D_FEAT + f], SCALE * v * xv);
                }
            }
        }
        __syncthreads();   // buf b free before it is overwritten next iter
        b ^= 1;
    }
}

// ---------------------------------------------------------------------------
// Kernel: out = ALPHA * h  (re-initializes the accumulator on every call,
// required because the harness does not re-poison between graph replays)
// ---------------------------------------------------------------------------
__global__ void init_out_kernel(const float* __restrict__ h,
                                float* __restrict__ out, int n)
{
    const int i4 = (blockIdx.x * blockDim.x + threadIdx.x) * 4;
    if (i4 + 3 < n) {
        const float4 v = *reinterpret_cast<const float4*>(h + i4);
        float4 o;
        o.x = ALPHA_F * v.x; o.y = ALPHA_F * v.y;
        o.z = ALPHA_F * v.z; o.w = ALPHA_F * v.w;
        *reinterpret_cast<float4*>(out + i4) = o;
    } else {
        // straight-line tail (n % 4 != 0 only in the generic case)
        if (i4     < n) out[i4]     = ALPHA_F * h[i4];
        if (i4 + 1 < n) out[i4 + 1] = ALPHA_F * h[i4 + 1];
        if (i4 + 2 < n) out[i4 + 2] = ALPHA_F * h[i4 + 2];
    }
}

// ---------------------------------------------------------------------------
extern "C" void kernel_launch(void* const* d_in, const int* in_sizes, int n_in,
                              void* d_out, int out_size, void* d_ws, size_t ws_size,
                              hipStream_t stream)
{
    const float* x    = (const float*)d_in[0];
    const float* h    = (const float*)d_in[1];
    const float* vals = (const float*)d_in[2];
    const int*   rows = (const int*)  d_in[3];
    const int*   cols = (const int*)  d_in[4];
    float*       out  = (float*)d_out;

    const int n_edges = in_sizes[2];

    // 1) out = ALPHA * h
    const int n4 = (out_size + 3) / 4;
    init_out_kernel<<<(n4 + 255) / 256, 256, 0, stream>>>(h, out, out_size);

    // 2) scatter-accumulate the weighted messages
    const int n_tiles = (n_edges + TILE_E - 1) / TILE_E;
    int grid = n_tiles < 2048 ? n_tiles : 2048;   // grid-stride => pipeline depth
    spmm_scatter_kernel<<<grid, BLOCK, 0, stream>>>(x, vals, rows, cols, out,
                                                    n_edges, n_tiles);
}